// FLinear2d_44968307589517
// MI455X (gfx1250) — compile-verified
//
#include <hip/hip_runtime.h>

// ---------------------------------------------------------------------------
// FLinear2d on MI455X (gfx1250), f32 end-to-end via V_WMMA_F32_16X16X4_F32.
//
//   B=64, CI=64, CO=128, SIN=4096, SOUT=1024
//   x  : [B, CI, SIN]   U : [CO, CI, SIN]   bU : [CO]
//   V  : [SOUT, CO, CI] bV: [SOUT]          out: [B, CO, SOUT]
//
// Stage 1 (per channel i):  u_i = X_i[64x4096] * U_i^T[4096x128] + bU
//   A/B tiles staged through LDS with GLOBAL_LOAD_ASYNC_TO_LDS_B128
//   (coalesced 256B rows, ASYNCcnt double buffering), WMMA fed by ds_load_b64.
//   Result stored transposed into w[o][b][i] ([CO, B, CI], 2 MB of d_ws).
// Stage 2 (per out-chan o): z_o = w_o[64x64] * V_o^T[64x1024] + bV
//   K=64 only; register-fed WMMA (V is read exactly once, w is L2-resident).
//
// Problem is HBM-bound (~268 MB unique traffic, ~5.4 GFLOP => ~11.5us at
// 23.3 TB/s); keep f32 precision, use WMMA f32 + async-LDS for coalescing.
// ---------------------------------------------------------------------------

typedef __attribute__((ext_vector_type(2))) float v2f;
typedef __attribute__((ext_vector_type(8))) float v8f;

#define BATCH 64
#define CI    64
#define CO    128
#define SIN   4096
#define SOUT  1024

#define KC    64            // K-chunk (floats) staged in LDS per buffer
#define NC    (SIN / KC)    // 64 chunks
#define LSTR  68            // padded LDS row stride: 272B = 16B-aligned,
                            // lanes land 4 banks apart -> conflict-free b64

// Generic pointer -> LDS byte offset (flat LDS addr truncates to offset).
__device__ __forceinline__ unsigned lds_off(const void* p) {
    return (unsigned)(unsigned long long)
        (const __attribute__((address_space(3))) char*)p;
}

// One coalesced 16-byte async copy global -> LDS (tracked by ASYNCcnt).
__device__ __forceinline__ void async_cp16(const float* g, float* l) {
    unsigned dst = lds_off(l);
    unsigned long long src = (unsigned long long)g;
    asm volatile("global_load_async_to_lds_b128 %0, %1, off"
                 :: "v"(dst), "v"(src) : "memory");
}

// ---------------------------------------------------------------------------
// Stage 1: grid (CI, CO/16), block 128 (4 waves; wave = M tile of batch).
// ---------------------------------------------------------------------------
__global__ __launch_bounds__(128)
void flin2d_stage1(const float* __restrict__ x,
                   const float* __restrict__ U,
                   const float* __restrict__ bU,
                   float* __restrict__ w)
{
    __shared__ float As[2][BATCH][LSTR];   // A chunk: 64 batch rows x KC
    __shared__ float Bs[2][16][LSTR];      // B chunk: 16 Co cols  x KC

    const int i      = blockIdx.x;         // input channel
    const int ntile  = blockIdx.y;         // Co tile
    const int tid    = threadIdx.x;
    const int wave   = tid >> 5;           // M tile 0..3
    const int lane   = tid & 31;
    const int lane16 = lane & 15;
    const int half   = lane >> 4;          // 0: K=0,1  1: K=2,3

    // Cooperative fill roles: 16 lanes x 16B cover one 256B row chunk.
    const int fr = tid >> 4;               // 0..7 (row-per-pass)
    const int fc = (tid & 15) * 4;         // float col within chunk

    const float* __restrict__ xch = x + (size_t)i * SIN;                       // + m*CI*SIN
    const float* __restrict__ Uch = U + (size_t)(ntile * 16) * (CI * SIN)
                                      + (size_t)i * SIN;                       // + n*CI*SIN

    // Issue the 10 async b128 loads that fill one buffer for chunk `c`.
    auto issue_fill = [&](int c, int buf) {
        const size_t kbase = (size_t)c * KC;
#pragma unroll
        for (int p = 0; p < 8; ++p) {      // A: 64 rows, 8 per pass
            const int r = p * 8 + fr;
            async_cp16(xch + (size_t)r * (CI * SIN) + kbase + fc, &As[buf][r][fc]);
        }
#pragma unroll
        for (int p = 0; p < 2; ++p) {      // B: 16 rows, 8 per pass
            const int n = p * 8 + fr;
            async_cp16(Uch + (size_t)n * (CI * SIN) + kbase + fc, &Bs[buf][n][fc]);
        }
    };

    v8f acc = {};
    issue_fill(0, 0);

    for (int c = 0; c < NC; ++c) {
        const int cur = c & 1;
        if (c + 1 < NC) {
            issue_fill(c + 1, cur ^ 1);
            // leave only the 10 just-issued copies outstanding
            asm volatile("s_wait_asynccnt 0xa" ::: "memory");
        } else {
            asm volatile("s_wait_asynccnt 0x0" ::: "memory");
        }
        __syncthreads();                   // buffer `cur` visible to all waves

        const float* At = &As[cur][wave * 16 + lane16][0];
        const float* Bt = &Bs[cur][lane16][0];
#pragma unroll
        for (int kk = 0; kk < KC; kk += 4) {
            const int k = kk + 2 * half;
            v2f a = *(const v2f*)(At + k); // A: (m, k..k+1)
            v2f b = *(const v2f*)(Bt + k); // B: (k..k+1, n)
            acc = __builtin_amdgcn_wmma_f32_16x16x4_f32(
                /*neg_a=*/false, a, /*neg_b=*/false, b,
                /*c_mod=*/(short)0, acc, /*reuse_a=*/false, /*reuse_b=*/false);
        }
        __syncthreads();                   // done consuming before refill
    }

    // D layout: VGPR v holds row M = wave*16 + v + 8*half, col N = n.
    const int n = ntile * 16 + lane16;
    const float bias = bU[n];
#pragma unroll
    for (int v = 0; v < 8; ++v) {
        const int bb = wave * 16 + v + 8 * half;   // batch index
        w[(size_t)n * (BATCH * CI) + (size_t)bb * CI + i] = acc[v] + bias;
    }
}

// ---------------------------------------------------------------------------
// Stage 2: grid (CO, SOUT/16), block 128 (4 waves; wave = M tile of batch).
// ---------------------------------------------------------------------------
__global__ __launch_bounds__(128)
void flin2d_stage2(const float* __restrict__ w,
                   const float* __restrict__ V,
                   const float* __restrict__ bV,
                   float* __restrict__ out)
{
    const int o      = blockIdx.x;         // output channel
    const int ntile  = blockIdx.y;         // Sout tile
    const int wave   = threadIdx.x >> 5;   // M tile 0..3
    const int lane   = threadIdx.x & 31;
    const int lane16 = lane & 15;
    const int half   = lane >> 4;

    const int m = wave  * 16 + lane16;     // batch row (A frag)
    const int t = ntile * 16 + lane16;     // Sout column (B frag)

    const float* __restrict__ Arow = w + (size_t)o * (BATCH * CI) + (size_t)m * CI;
    const float* __restrict__ Brow = V + (size_t)t * (CO * CI) + (size_t)o * CI;

    v8f acc = {};
#pragma unroll
    for (int k0 = 0; k0 < CI; k0 += 4) {
        const int k = k0 + 2 * half;
        v2f a = *(const v2f*)(Arow + k);
        v2f b = *(const v2f*)(Brow + k);
        acc = __builtin_amdgcn_wmma_f32_16x16x4_f32(
            false, a, false, b, (short)0, acc, false, false);
    }

    const float bias = bV[t];
#pragma unroll
    for (int v = 0; v < 8; ++v) {
        const int bb = wave * 16 + v + 8 * half;   // batch index
        out[(size_t)bb * (CO * SOUT) + (size_t)o * SOUT + t] = acc[v] + bias;
    }
}

extern "C" void kernel_launch(void* const* d_in, const int* in_sizes, int n_in,
                              void* d_out, int out_size, void* d_ws, size_t ws_size,
                              hipStream_t stream)
{
    const float* x  = (const float*)d_in[0];   // [B, CI, SIN]
    const float* U  = (const float*)d_in[1];   // [CO, CI, SIN]
    const float* bU = (const float*)d_in[2];   // [CO]
    const float* V  = (const float*)d_in[3];   // [SOUT, CO, CI]
    const float* bV = (const float*)d_in[4];   // [SOUT]
    float* out = (float*)d_out;                // [B, CO, SOUT]
    float* w   = (float*)d_ws;                 // [CO, B, CI] = 2 MB scratch

    dim3 grid1(CI, CO / 16);                   // (64, 8)
    flin2d_stage1<<<grid1, 128, 0, stream>>>(x, U, bU, w);

    dim3 grid2(CO, SOUT / 16);                 // (128, 64)
    flin2d_stage2<<<grid2, 128, 0, stream>>>(w, V, bV, out);
}